// Conv2d_82162724372967
// MI455X (gfx1250) — compile-verified
//
#include <hip/hip_runtime.h>
#include <stdint.h>

// Conv2d 3x3, Cin=Cout=4, 4096x4096, fp32, pad=1, stride=1.
// Implicit GEMM on V_WMMA_F32_16X16X4_F32 (M=16 pixels, N=cout(4 of 16), K=cin(4) x 9 taps).
// Memory-bound (~537 MB @ 23.3 TB/s => ~23us floor):
//  - LDS tile + halo, each HBM byte read ~once (10% halo overhead)
//  - interior blocks stage via GLOBAL_LOAD_ASYNC_TO_LDS_B32 (ASYNCcnt path, no VGPR round trip)
//  - border blocks use bounds-checked sync loads for zero padding
//  - output stores are non-temporal b128 to keep input lines resident in the 192MB L2

#define CIN 4
#define IMH 4096
#define IMW 4096
#define HWSZ (IMH * IMW)
#define TR 32            // output rows per block
#define TC 64            // output cols per block
#define LR (TR + 2)      // 34 halo rows
#define LC (TC + 2)      // 66 halo cols

typedef float v2f __attribute__((ext_vector_type(2)));
typedef float v4f __attribute__((ext_vector_type(4)));
typedef float v8f __attribute__((ext_vector_type(8)));

__global__ __launch_bounds__(256) void conv3x3_wmma_f32(
    const float* __restrict__ x,     // [4, 4096, 4096]
    const float* __restrict__ ker,   // [4, 4, 3, 3]
    float* __restrict__ out)         // [4, 4096, 4096]
{
    // LDS layout: [r][c][ci], ci innermost -> one lane's A fragment is one aligned ds_load_b64.
    __shared__ float lds[LR * LC * CIN];   // 35904 B

    const int tid  = threadIdx.x;
    const int lane = tid & 31;
    const int wv   = tid >> 5;

    const int col0 = blockIdx.x * TC;
    const int row0 = blockIdx.y * TR;

    const int total = LR * LC * CIN;

    // Whole halo in-bounds? (true for ~95% of blocks)
    const bool interior = (row0 >= 1) && (row0 + TR + 1 <= IMH) &&
                          (col0 >= 1) && (col0 + TC + 1 <= IMW);

    if (interior) {
        // ---- Async DMA staging: global -> LDS directly, tracked by ASYNCcnt ----
        for (int e = tid; e < total; e += 256) {
            int ci  = e / (LR * LC);
            int rem = e - ci * (LR * LC);
            int r   = rem / LC;
            int c   = rem - r * LC;
            unsigned ldsoff = (unsigned)(uintptr_t)(&lds[(r * LC + c) * CIN + ci]);
            int goff = (int)(((unsigned)(ci * HWSZ) +
                              (unsigned)((row0 + r - 1) * IMW + (col0 + c - 1))) * 4u);
            // GVS mode: mem = SADDR(64b) + VADDR(i32) ; LDS dest byte addr in VDST
            asm volatile("global_load_async_to_lds_b32 %0, %1, %2"
                         :: "v"(ldsoff), "v"(goff), "s"(x)
                         : "memory");
        }
        asm volatile("s_wait_asynccnt 0x0" ::: "memory");
    } else {
        // ---- Border blocks: bounds-checked sync staging with zero fill ----
        for (int e = tid; e < total; e += 256) {
            int ci  = e / (LR * LC);
            int rem = e - ci * (LR * LC);
            int r   = rem / LC;
            int c   = rem - r * LC;
            int gh  = row0 + r - 1;
            int gw  = col0 + c - 1;
            float v = 0.0f;
            if ((unsigned)gh < (unsigned)IMH && (unsigned)gw < (unsigned)IMW)
                v = x[ci * HWSZ + gh * IMW + gw];
            lds[(r * LC + c) * CIN + ci] = v;
        }
    }
    __syncthreads();

    // ---- Per-lane B fragments: B[k][n] = ker[n][k][kh][kw]  (4x16, n>=4 zero) ----
    // B 4x16 f32 layout (mirrors A): VGPR0 = K rows 0|2 by lane half, VGPR1 = K rows 1|3.
    const int n  = lane & 15;
    const int k0 = (lane < 16) ? 0 : 2;
    v2f bt[9];
#pragma unroll
    for (int kh = 0; kh < 3; ++kh)
#pragma unroll
        for (int kw = 0; kw < 3; ++kw) {
            v2f b;
            b[0] = 0.0f;
            b[1] = 0.0f;
            if (n < CIN) {
                b[0] = ker[((n * CIN + k0    ) * 3 + kh) * 3 + kw];
                b[1] = ker[((n * CIN + k0 + 1) * 3 + kh) * 3 + kw];
            }
            bt[kh * 3 + kw] = b;
        }

    const int m = lane & 15;   // A-matrix row = output pixel within 16-col segment

    // ---- Each wave: 1-row x 16-col output tiles, all 4 couts per WMMA chain ----
    for (int t = wv; t < TR * (TC / 16); t += 8) {
        const int r   = t >> 2;         // output row within block tile
        const int wc0 = (t & 3) * 16;   // col segment start within block tile

        v8f acc = {};
#pragma unroll
        for (int kh = 0; kh < 3; ++kh)
#pragma unroll
            for (int kw = 0; kw < 3; ++kw) {
                const float* p = &lds[((r + kh) * LC + (wc0 + m + kw)) * CIN + k0];
                v2f a = *(const v2f*)p;   // aligned 8B ds_load_b64
                acc = __builtin_amdgcn_wmma_f32_16x16x4_f32(
                    /*neg_a=*/false, a, /*neg_b=*/false, bt[kh * 3 + kw],
                    /*c_mod=*/(short)0, acc, /*reuse_a=*/false, /*reuse_b=*/false);
            }

        // D layout: VGPR v of lane L holds D[M = v + (L<16?0:8)][N = L%16].
        if (n < CIN) {
            const int mb = (lane < 16) ? 0 : 8;
            float* o = &out[n * HWSZ + (row0 + r) * IMW + (col0 + wc0 + mb)];
            v4f lo = { acc[0], acc[1], acc[2], acc[3] };
            v4f hi = { acc[4], acc[5], acc[6], acc[7] };
            __builtin_nontemporal_store(lo, (v4f*)(o));       // NT b128: don't pollute L2
            __builtin_nontemporal_store(hi, (v4f*)(o + 4));
        }
    }
}

extern "C" void kernel_launch(void* const* d_in, const int* in_sizes, int n_in,
                              void* d_out, int out_size, void* d_ws, size_t ws_size,
                              hipStream_t stream) {
    const float* x   = (const float*)d_in[0];   // [4,4096,4096] fp32
    const float* ker = (const float*)d_in[1];   // [4,4,3,3] fp32
    float* out = (float*)d_out;                 // [4,4096,4096] fp32

    dim3 grid(IMW / TC, IMH / TR);   // 64 x 128 blocks
    conv3x3_wmma_f32<<<grid, 256, 0, stream>>>(x, ker, out);
}